// FNOBlock_73830487818863
// MI455X (gfx1250) — compile-verified
//
#include <hip/hip_runtime.h>
#include <hip/hip_bf16.h>

// FNO block on MI455X (gfx1250), wave32, WMMA f32 16x16x4 everywhere.
// B=16, C=64, H=256, W=256, M=16. All spectral stages are partial DFTs
// implemented as f32 WMMA GEMMs; 1x1 conv is fused into the final
// inverse-W stage together with bias + exact GELU.
//
// LDS-staged B matrices use a k-paired interleaved layout so each WMMA
// B fragment (B[k][n], B[k+1][n]) is contiguous -> single ds_load_b64,
// no v_mov repacking in the inner loops.

typedef __attribute__((ext_vector_type(2))) float v2f;
typedef __attribute__((ext_vector_type(8))) float v8f;

__device__ __forceinline__ v8f wmma4(v2f a, v2f b, v8f c) {
  // D = A(16x4) * B(4x16) + C, f32 in/out
  return __builtin_amdgcn_wmma_f32_16x16x4_f32(
      false, a, false, b, (short)0, c, false, false);
}

__device__ __forceinline__ v2f ld2(const float* p) {
  return *(const v2f*)p;  // 8B-aligned by construction (even float offsets)
}

// Store one 16x16 f32 C/D tile. Lanes 0-15 hold M=r, lanes 16-31 M=8+r.
__device__ __forceinline__ void store_tile(float* C, long sCm, long sCn,
                                           int lane, v8f acc) {
  int half = lane >> 4, lr = lane & 15;
  float* Cp = C + (long)(8 * half) * sCm + (long)lr * sCn;
#pragma unroll
  for (int r = 0; r < 8; ++r) Cp[(long)r * sCm] = acc[r];
}

// ---------------------------------------------------------------------------
// Twiddle-table init (deterministic; runs every call; graph-safe).
// ---------------------------------------------------------------------------
__global__ void k_init(float* Fw, float* FhcR, float* FhcI, float* IFhR,
                       float* IFhI, float* IFw) {
  const float TWO_PI_256 = 6.283185307179586f / 256.0f;
  int t = blockIdx.x * blockDim.x + threadIdx.x;  // 0..16383
  if (t < 8192) {  // Fw [w][col]: cols 0-15 cos, 16-31 -sin
    int w = t >> 5, col = t & 31, j = col & 15;
    float th = TWO_PI_256 * (float)((w * j) & 255);
    Fw[t] = (col < 16) ? cosf(th) : -sinf(th);
  }
  {  // Fhc [kh][512]: R=[cos|sin], I=[-sin|cos]
    int kh = t >> 9, k = t & 511, h = k & 255;
    int f = (kh < 16) ? kh : kh + 224;
    float th = TWO_PI_256 * (float)((f * h) & 255);
    float c = cosf(th), s = sinf(th);
    FhcR[t] = (k < 256) ? c : s;
    FhcI[t] = (k < 256) ? -s : c;
  }
  {  // IFh [h][64]: R=[cos|-sin], I=[sin|cos]
    int h = t >> 6, k = t & 63, kh = k & 31;
    int f = (kh < 16) ? kh : kh + 224;
    float th = TWO_PI_256 * (float)((f * h) & 255);
    float c = cosf(th), s = sinf(th);
    IFhR[t] = (k < 32) ? c : -s;
    IFhI[t] = (k < 32) ? s : c;
  }
  if (t < 8192) {  // IFw [k][w]: rows 0-15 c*cos/65536, 16-31 -c*sin/65536
    int k = t >> 8, w = t & 255, kw = k & 15;
    float th = TWO_PI_256 * (float)((kw * w) & 255);
    float sc = ((kw == 0) ? 1.0f : 2.0f) * (1.0f / 65536.0f);
    IFw[t] = (k < 16) ? cosf(th) * sc : -sinf(th) * sc;
  }
}

// ---------------------------------------------------------------------------
// Stage A: forward W-DFT. rows=(b,c,h) (262144), K=256. Output
// tmpW[B][C][2][H][16]. Fw staged in LDS with k-paired layout.
// ---------------------------------------------------------------------------
__global__ void k_fwd_w(const float* __restrict__ x,
                        const float* __restrict__ Fw,
                        float* __restrict__ tmpW) {
  extern __shared__ float sFw[];  // 8192 floats, paired: (k>>1)*64+n*2+(k&1)
  for (int i = threadIdx.x; i < 8192; i += blockDim.x) {
    int k = i >> 5, n = i & 31;
    sFw[(k >> 1) * 64 + n * 2 + (k & 1)] = Fw[i];
  }
  __syncthreads();
  int wave = threadIdx.x >> 5, lane = threadIdx.x & 31;
  int half = lane >> 4, lr = lane & 15;
  long r0 = ((long)blockIdx.x * 4 + wave) * 16;
  const float* Ap = x + (r0 + lr) * 256 + 2 * half;
  const float* BR = sFw + half * 64 + lr * 2;  // pair (k+2h, k+2h+1) at col lr
  const float* BI = BR + 32;                   // col 16+lr
  v8f accR = {}, accI = {};
  for (int k = 0; k < 256; k += 4) {
    v2f a = ld2(Ap);
    v2f bR = ld2(BR);
    v2f bI = ld2(BI);
    accR = wmma4(a, bR, accR);
    accI = wmma4(a, bI, accI);
    Ap += 4; BR += 128; BI += 128;
  }
  long bc = r0 >> 8;
  int h0 = (int)(r0 & 255);
  float* CR = tmpW + bc * 8192 + (long)h0 * 16;  // plane 0 (R)
  store_tile(CR, 16, 1, lane, accR);
  store_tile(CR + 4096, 16, 1, lane, accI);      // plane 1 (I)
}

// ---------------------------------------------------------------------------
// Stage B: forward H-DFT. Per (b,c): M=32,N=16,K=512 (R/I planes stacked).
// tmpW slice staged in LDS with k-paired layout.
// ---------------------------------------------------------------------------
__global__ void k_fwd_h(const float* __restrict__ tmpW,
                        const float* __restrict__ FhcR,
                        const float* __restrict__ FhcI,
                        float* __restrict__ XftR, float* __restrict__ XftI) {
  extern __shared__ float sB[];  // 8192 floats, paired: (k>>1)*32+n*2+(k&1)
  long bc = blockIdx.x;
  const float* src = tmpW + bc * 8192;
  for (int i = threadIdx.x; i < 8192; i += blockDim.x) {
    int k = i >> 4, n = i & 15;
    sB[(k >> 1) * 32 + n * 2 + (k & 1)] = src[i];
  }
  __syncthreads();
  int wave = threadIdx.x >> 5, lane = threadIdx.x & 31;
  int mt = wave & 1, ri = wave >> 1;  // kh tile, R/I output select
  int half = lane >> 4, lr = lane & 15;
  const float* Apt = ((ri ? FhcI : FhcR) + (long)mt * 16 * 512) +
                     (long)lr * 512 + 2 * half;
  const float* Bpt = sB + half * 32 + lr * 2;
  v8f acc = {};
  for (int k = 0; k < 512; k += 4) {
    v2f a = ld2(Apt);
    v2f b = ld2(Bpt);
    acc = wmma4(a, b, acc);
    Apt += 4; Bpt += 64;
  }
  int b = (int)(bc >> 6), c = (int)(bc & 63);
  float* C = (ri ? XftI : XftR) + (long)mt * 16 * 16384 + b * 64 + c;
  store_tile(C, 16384, 1024, lane, acc);  // sCm over kh, sCn over kw
}

// ---------------------------------------------------------------------------
// Stage C: per-mode complex channel mix. 512 modes; per mode a complex
// 16(b) x 64(i) x 64(o) GEMM: YR = XR*WR - XI*WI ; YI = XR*WI + XI*WR.
// ---------------------------------------------------------------------------
__global__ void k_modes(const float* __restrict__ XftR,
                        const float* __restrict__ XftI,
                        const float* __restrict__ w1r,
                        const float* __restrict__ w1i,
                        const float* __restrict__ w2r,
                        const float* __restrict__ w2i,
                        float* __restrict__ YR, float* __restrict__ YI) {
  int mode = blockIdx.x;           // kh*16 + kw
  int kh = mode >> 4, kw = mode & 15;
  int wave = threadIdx.x >> 5, lane = threadIdx.x & 31;
  int half = lane >> 4, lr = lane & 15;
  int xx = (kh < 16) ? kh : kh - 16;
  long woff = (long)xx * 16 + kw;  // w[i][o][x][y]: +(i*64+o)*256
  const float* wr = ((kh < 16) ? w1r : w2r) + woff;
  const float* wi = ((kh < 16) ? w1i : w2i) + woff;
  const float* ApR = XftR + (long)mode * 1024 + lr * 64 + 2 * half;
  const float* ApI = XftI + (long)mode * 1024 + lr * 64 + 2 * half;
  const float* BpR = wr + (long)(wave * 16 + lr) * 256 + (long)(2 * half) * 16384;
  const float* BpI = wi + (long)(wave * 16 + lr) * 256 + (long)(2 * half) * 16384;
  v8f accR = {}, accI = {};
  for (int k = 0; k < 64; k += 4) {
    v2f aR = ld2(ApR);
    v2f aI = ld2(ApI);
    v2f bR;  bR.x = BpR[0];  bR.y = BpR[16384];
    v2f bI;  bI.x = BpI[0];  bI.y = BpI[16384];
    v2f naI; naI.x = -aI.x;  naI.y = -aI.y;
    accR = wmma4(aR, bR, accR);
    accR = wmma4(naI, bI, accR);
    accI = wmma4(aR, bI, accI);
    accI = wmma4(aI, bR, accI);
    ApR += 4; ApI += 4; BpR += 4 * 16384; BpI += 4 * 16384;
  }
  store_tile(YR + (long)mode * 1024 + wave * 16, 64, 1, lane, accR);
  store_tile(YI + (long)mode * 1024 + wave * 16, 64, 1, lane, accI);
}

// ---------------------------------------------------------------------------
// Stage D: inverse H-DFT. Per (b,o): M=256(h), N=16(kw), K=64 (YR|YI
// stacked). Y slice staged in LDS with k-paired layout. Output
// tmp2[B][C][H][32] (per-h interleaved R/I) so stage E's A is affine.
// ---------------------------------------------------------------------------
__global__ void k_inv_h(const float* __restrict__ Y,  // YR base; YI = +524288
                        const float* __restrict__ IFhR,
                        const float* __restrict__ IFhI,
                        float* __restrict__ tmp2) {
  extern __shared__ float sY[];  // 1024 floats, paired: (k>>1)*32+n*2+(k&1)
  int bo = blockIdx.x;
  int b = bo >> 6, o = bo & 63;
  const float* src = Y + (long)b * 64 + o;
  for (int i = threadIdx.x; i < 1024; i += blockDim.x) {
    int k = i >> 4, n = i & 15;
    sY[(k >> 1) * 32 + n * 2 + (k & 1)] = src[(long)k * 16384 + (long)n * 1024];
  }
  __syncthreads();
  int widx = blockIdx.y * 4 + (threadIdx.x >> 5);
  int ri = widx >> 4, mt = widx & 15;
  int lane = threadIdx.x & 31, half = lane >> 4, lr = lane & 15;
  const float* Apt = ((ri ? IFhI : IFhR) + (long)mt * 16 * 64) +
                     (long)lr * 64 + 2 * half;
  const float* Bpt = sY + half * 32 + lr * 2;
  v8f acc = {};
  for (int k = 0; k < 64; k += 4) {
    v2f a = ld2(Apt);
    v2f b = ld2(Bpt);
    acc = wmma4(a, b, acc);
    Apt += 4; Bpt += 64;
  }
  float* C = tmp2 + (long)bo * 8192 + (long)(mt * 16) * 32 + ri * 16;
  store_tile(C, 32, 1, lane, acc);
}

// ---------------------------------------------------------------------------
// Stage E (fused): per (b,h) row: out[o,w] = gelu(Wc@x + tmp2@IFw + bc).
// Two WMMA accumulations into one accumulator kill the res round-trip.
// Wc (A, row-major), IFw (B, k-paired), tmp2 slice (A, row-major), bc in
// LDS (~56.5 KB). x streamed from global (the unavoidable second read).
// ---------------------------------------------------------------------------
__global__ void k_final(const float* __restrict__ x,
                        const float* __restrict__ Wc,
                        const float* __restrict__ bcv,
                        const float* __restrict__ IFw,
                        const float* __restrict__ tmp2,
                        float* __restrict__ out) {
  extern __shared__ float sm[];
  float* sWc = sm;            // 4096 (row-major [o][c], A operand)
  float* sIFw = sm + 4096;    // 8192 (k-paired B operand)
  float* sT2 = sm + 12288;    // 2048 (row-major [o][32], A operand)
  float* sBc = sm + 14336;    // 64
  int bh = blockIdx.x;
  int b = bh >> 8, h = bh & 255;
  for (int i = threadIdx.x; i < 4096; i += 128) sWc[i] = Wc[i];
  for (int i = threadIdx.x; i < 8192; i += 128) {
    int k = i >> 8, w = i & 255;
    sIFw[(k >> 1) * 512 + w * 2 + (k & 1)] = IFw[i];
  }
  const float* t2src = tmp2 + (long)b * 524288 + (long)h * 32;
  for (int i = threadIdx.x; i < 2048; i += 128) {
    int o = i >> 5, k = i & 31;
    sT2[i] = t2src[(long)o * 8192 + k];
  }
  if (threadIdx.x < 64) sBc[threadIdx.x] = bcv[threadIdx.x];
  __syncthreads();
  int wave = threadIdx.x >> 5, lane = threadIdx.x & 31;
  int half = lane >> 4, lr = lane & 15;
  int o0 = wave * 16;
  const float* xb = x + (long)b * 4194304 + (long)h * 256;  // +c*65536+w
  for (int wt = 0; wt < 16; ++wt) {
    int w0 = wt * 16;
    v8f acc = {};
    // conv branch: A = Wc[o][c] (LDS v2f), B = x[c][w] (global, K=64)
    const float* Apt = sWc + (o0 + lr) * 64 + 2 * half;
    const float* Bpt = xb + (long)(2 * half) * 65536 + w0 + lr;
    for (int k = 0; k < 64; k += 4) {
      v2f a = ld2(Apt);
      v2f bb; bb.x = Bpt[0]; bb.y = Bpt[65536];
      acc = wmma4(a, bb, acc);
      Apt += 4; Bpt += 4L * 65536;
    }
    // spectral branch: A = tmp2[o][k] (LDS v2f), B = IFw k-paired (K=32)
    const float* Apt2 = sT2 + (o0 + lr) * 32 + 2 * half;
    const float* Bpt2 = sIFw + half * 512 + (w0 + lr) * 2;
    for (int k = 0; k < 32; k += 4) {
      v2f a = ld2(Apt2);
      v2f bb = ld2(Bpt2);
      acc = wmma4(a, bb, acc);
      Apt2 += 4; Bpt2 += 1024;
    }
    // epilogue: bias + exact gelu, write out once
    float* Cp = out + ((long)(b * 64 + o0 + 8 * half) * 256 + h) * 256 + w0 + lr;
#pragma unroll
    for (int r = 0; r < 8; ++r) {
      float v = acc[r] + sBc[o0 + 8 * half + r];
      float g = 0.5f * v * (1.0f + erff(v * 0.7071067811865476f));
      Cp[(long)r * 65536] = g;
    }
  }
}

// ---------------------------------------------------------------------------
extern "C" void kernel_launch(void* const* d_in, const int* in_sizes, int n_in,
                              void* d_out, int out_size, void* d_ws,
                              size_t ws_size, hipStream_t stream) {
  const float* x   = (const float*)d_in[0];
  const float* Wc  = (const float*)d_in[1];
  const float* bc  = (const float*)d_in[2];
  const float* w1r = (const float*)d_in[3];
  const float* w1i = (const float*)d_in[4];
  const float* w2r = (const float*)d_in[5];
  const float* w2i = (const float*)d_in[6];
  float* out = (float*)d_out;
  float* ws = (float*)d_ws;

  // Workspace layout (floats); total ~18.96M floats (~76 MB).
  float* Fw   = ws;             // 8192
  float* FhcR = ws + 8192;      // 16384
  float* FhcI = ws + 24576;     // 16384
  float* IFhR = ws + 40960;     // 16384
  float* IFhI = ws + 57344;     // 16384
  float* IFw  = ws + 73728;     // 8192
  float* tmpW = ws + 81920;     // 8388608  [B][C][2][H][16]
  float* XftR = ws + 8470528;   // 524288   [mode][b][c]
  float* XftI = ws + 8994816;   // 524288
  float* YR   = ws + 9519104;   // 524288   [mode][b][o]
  float* YI   = ws + 10043392;  // 524288   (contiguous after YR)
  float* tmp2 = ws + 10567680;  // 8388608  [B][C][H][32]

  k_init<<<64, 256, 0, stream>>>(Fw, FhcR, FhcI, IFhR, IFhI, IFw);
  k_fwd_w<<<4096, 128, 8192 * 4, stream>>>(x, Fw, tmpW);
  k_fwd_h<<<1024, 128, 8192 * 4, stream>>>(tmpW, FhcR, FhcI, XftR, XftI);
  k_modes<<<512, 128, 0, stream>>>(XftR, XftI, w1r, w1i, w2r, w2i, YR, YI);
  dim3 gD(1024, 8);
  k_inv_h<<<gD, 128, 1024 * 4, stream>>>(YR, IFhR, IFhI, tmp2);
  k_final<<<4096, 128, 14400 * 4, stream>>>(x, Wc, bc, IFw, tmp2, out);
}